// HenonLayer_63299228009060
// MI455X (gfx1250) — compile-verified
//
#include <hip/hip_runtime.h>
#include <stdint.h>

#define NI     2048
#define NSTEPS 4

typedef uint32_t u32;
typedef u32   u32x4 __attribute__((ext_vector_type(4)));
typedef u32   u32x8 __attribute__((ext_vector_type(8)));
typedef float v2f   __attribute__((ext_vector_type(2)));
typedef float v8f   __attribute__((ext_vector_type(8)));

// ---- gfx1250 hardware tanh -------------------------------------------------
__device__ __forceinline__ float fast_tanhf(float x) {
#if __has_builtin(__builtin_amdgcn_tanhf)
    return __builtin_amdgcn_tanhf(x);          // v_tanh_f32
#elif __has_builtin(__builtin_amdgcn_tanh_f32)
    return __builtin_amdgcn_tanh_f32(x);
#else
    return tanhf(x);
#endif
}

__device__ __forceinline__ void wait_tensorcnt0() {
#if __has_builtin(__builtin_amdgcn_s_wait_tensorcnt)
    __builtin_amdgcn_s_wait_tensorcnt(0);
#else
    asm volatile("s_wait_tensorcnt 0x0" ::: "memory");
#endif
}

// ---- TDM: 1-D f32 tile, global -> LDS (CDNA5 ISA §8.3/§8.4 descriptor) -----
__device__ __forceinline__ void tdm_load_f32_1d(u32 lds_byte_off,
                                                const void* gptr, u32 n) {
    const uint64_t ga = (uint64_t)(uintptr_t)gptr;
    u32x4 g0;
    g0.x = 1u;                                            // count=1, user D#
    g0.y = lds_byte_off;                                  // lds_addr (bytes)
    g0.z = (u32)(ga & 0xFFFFFFFFu);                       // global_addr[31:0]
    g0.w = (u32)((ga >> 32) & 0x01FFFFFFu) | (2u << 30);  // ga[56:32] | type=2

    u32x8 g1;
    g1.s0 = (2u << 16);                 // workgroup_mask=0, data_size=2 (4 B)
    g1.s1 = (n & 0xFFFFu) << 16;        // tensor_dim0[15:0]
    g1.s2 = ((n >> 16) & 0xFFFFu)       // tensor_dim0[31:16]
          | (1u << 16);                 // tensor_dim1 = 1
    g1.s3 = (n & 0xFFFFu) << 16;        // tile_dim0 = n
    g1.s4 = 1u;                         // tile_dim1 = 1
    g1.s5 = n;                          // tensor_dim0_stride
    g1.s6 = 0u;
    g1.s7 = 0u;

    asm volatile("tensor_load_to_lds %0, %1" :: "s"(g0), "s"(g1) : "memory");
}

// ---- one sweep over the hidden layer ---------------------------------------
// Wave covers 16 rows. Lane L: row m=L&15, K-half h=L>>4. Per 4-K chunk the
// lane supplies A = {tanh^2(s*Win[k]+bin[k])} at k = kc+2h, kc+2h+1 (the f32
// 16x4 A layout) and B = w2 pair at the same k (valid since all B columns are
// identical). v_wmma_f32_16x16x4_f32 accumulates 16x16 f32; column value for
// row m is Sum_k q[m,k]*w2[k]. If FIRST, also harvest Sum w2 (half per lane
// half, combined by the caller via shfl_xor 16).
template <bool FIRST>
__device__ __forceinline__ v8f sweep(const v2f* __restrict__ winp,
                                     const v2f* __restrict__ binp,
                                     const v2f* __restrict__ w2p,
                                     int h, float s, float& cw)
{
    v8f acc = {};
    float c0 = 0.f, c1 = 0.f;
    #pragma unroll 2
    for (int i = 0; i < NI / 4; ++i) {
        const int idx = 2 * i + h;          // ds_load_b64, half-wave broadcast
        v2f w = winp[idx];
        v2f b = binp[idx];
        v2f c = w2p[idx];
        float t0 = fast_tanhf(fmaf(s, w.x, b.x));
        float t1 = fast_tanhf(fmaf(s, w.y, b.y));
        v2f a;
        a.x = t0 * t0;
        a.y = t1 * t1;
        acc = __builtin_amdgcn_wmma_f32_16x16x4_f32(
                  false, a, false, c, (short)0, acc, false, false);
        if (FIRST) { c0 += c.x; c1 += c.y; }
    }
    if (FIRST) cw = c0 + c1;
    return acc;
}

// Extract row m's accumulated value for lane L from the 16x16 f32 D layout
// (lane L holds rows 8h..8h+7 in acc[0..7]; all columns identical).
__device__ __forceinline__ float extract_row(v8f acc, int lane)
{
    const bool b0 = (lane & 1) != 0;
    const bool b1 = (lane & 2) != 0;
    const bool b2 = (lane & 4) != 0;
    float x0 = b0 ? acc[1] : acc[0];
    float x1 = b0 ? acc[3] : acc[2];
    float x2 = b0 ? acc[5] : acc[4];
    float x3 = b0 ? acc[7] : acc[6];
    float y0 = b1 ? x1 : x0;
    float y1 = b1 ? x3 : x2;
    float cand = b2 ? y1 : y0;                    // row (L&7) + 8*(L>>4)
    float other = __shfl_xor(cand, 16, 32);       // partner half-wave
    const int lane_swap = ((lane >> 3) ^ (lane >> 4)) & 1;
    return lane_swap ? other : cand;              // row L&15
}

// ---------------------------------------------------------------------------
__global__ __launch_bounds__(256)
void henon_kernel(const float* __restrict__ z,
                  const float* __restrict__ Win,
                  const float* __restrict__ Wout,
                  const float* __restrict__ bin,
                  const float* __restrict__ eta,
                  const float* __restrict__ ymean,
                  const float* __restrict__ ydiam,
                  float* __restrict__ out, int B)
{
    __shared__ __align__(16) float sm[3 * NI];
    float* __restrict__ sWin = sm;
    float* __restrict__ sBin = sm + NI;
    float* __restrict__ sW2  = sm + 2 * NI;

    const int tid = threadIdx.x;

    // Wave 0 stages Win and bin into LDS via the Tensor Data Mover.
    if (tid < 32) {
        const u32 base = (u32)(uintptr_t)sm;   // flat LDS addr[31:0] = offset
        tdm_load_f32_1d(base,          Win, NI);
        tdm_load_f32_1d(base + NI * 4, bin, NI);
        wait_tensorcnt0();
    }
    __syncthreads();

    const float inv_ydiam = 1.0f / ydiam[0];

    for (int i = tid; i < NI; i += 256)
        sW2[i] = sWin[i] * Wout[i] * inv_ydiam;
    __syncthreads();

    const int lane = tid & 31;
    const int wave = tid >> 5;
    const int m    = lane & 15;
    const int h    = lane >> 4;                // K-half of this lane

    int row = blockIdx.x * 128 + wave * 16 + m;   // 8 waves * 16 rows / block
    const bool writer = (h == 0) && (row < B);
    if (row >= B) row = B - 1;                 // clamp: keep EXEC all-1 (WMMA)

    const float e  = eta[0];
    const float ym = ymean[0];

    float X = z[2 * row + 0];
    float Y = z[2 * row + 1];

    const v2f* __restrict__ winp = (const v2f*)sWin;
    const v2f* __restrict__ binp = (const v2f*)sBin;
    const v2f* __restrict__ w2p  = (const v2f*)sW2;

    // ---- step 0 (peeled): also harvest w2sum ------------------------------
    float w2sum;
    {
        const float s = (Y - ym) * inv_ydiam;
        float cw;
        v8f acc = sweep<true>(winp, binp, w2p, h, s, cw);
        w2sum = cw + __shfl_xor(cw, 16, 32);   // both K-halves combined
        const float grad = w2sum - extract_row(acc, lane);
        const float Xn = Y + e;
        const float Yn = grad - X;
        X = Xn; Y = Yn;
    }

    // ---- steps 1..3 -------------------------------------------------------
    #pragma unroll 1
    for (int step = 1; step < NSTEPS; ++step) {
        const float s = (Y - ym) * inv_ydiam;
        float cw;
        v8f acc = sweep<false>(winp, binp, w2p, h, s, cw);
        const float grad = w2sum - extract_row(acc, lane);
        const float Xn = Y + e;
        const float Yn = grad - X;
        X = Xn; Y = Yn;
    }

    if (writer) {
        out[2 * row + 0] = X;
        out[2 * row + 1] = Y;
    }
}

// ---------------------------------------------------------------------------
extern "C" void kernel_launch(void* const* d_in, const int* in_sizes, int n_in,
                              void* d_out, int out_size, void* d_ws, size_t ws_size,
                              hipStream_t stream)
{
    const float* z     = (const float*)d_in[0];
    const float* Win   = (const float*)d_in[1];
    const float* Wout  = (const float*)d_in[2];
    const float* bin   = (const float*)d_in[3];
    const float* eta   = (const float*)d_in[4];
    const float* ymean = (const float*)d_in[5];
    const float* ydiam = (const float*)d_in[6];
    float* out = (float*)d_out;

    const int B = in_sizes[0] / 2;             // 65536 rows
    const int rows_per_block = 128;            // 8 waves * 16 rows
    const int blocks = (B + rows_per_block - 1) / rows_per_block;

    henon_kernel<<<blocks, 256, 0, stream>>>(z, Win, Wout, bin, eta,
                                             ymean, ydiam, out, B);
}